// SGNN_11235634446461
// MI455X (gfx1250) — compile-verified
//
#include <hip/hip_runtime.h>

#define N_NODES 250000
#define N_EDGES 1000000
#define N_SUB   1000
#define DIM     64
#define FIN     16
#define FE      8
#define NLAYER  3
#define TS      68   // LDS tile row stride (floats); 68%4==0 -> rows stay 16B aligned

typedef __attribute__((ext_vector_type(2))) float v2f;
typedef __attribute__((ext_vector_type(4))) float v4f;
typedef __attribute__((ext_vector_type(8))) float v8f;

// ---------------------------------------------------------------------------
// utility kernels
// ---------------------------------------------------------------------------
__global__ void zero_kernel(float* __restrict__ p, long long n) {
    long long i = (long long)blockIdx.x * blockDim.x + threadIdx.x;
    long long stride = (long long)gridDim.x * blockDim.x;
    for (; i < n; i += stride) p[i] = 0.0f;
}

__global__ void count_kernel(const int* __restrict__ node_idx, float* __restrict__ cnt) {
    int i = blockIdx.x * blockDim.x + threadIdx.x;
    if (i < N_NODES) unsafeAtomicAdd(&cnt[node_idx[i]], 1.0f);
}

// dst[seg[i]][c] += src[i][c]   over (i,c) in [N,64]
__global__ void segsum_kernel(const float* __restrict__ src, const int* __restrict__ seg,
                              float* __restrict__ dst) {
    int idx = blockIdx.x * blockDim.x + threadIdx.x;
    if (idx < N_NODES * DIM) {
        int i = idx >> 6, c = idx & 63;
        unsafeAtomicAdd(&dst[seg[i] * DIM + c], src[idx]);
    }
}

// h = x @ W_init + b_init   (x: [N,16], W: [16,64])
__global__ __launch_bounds__(256) void init_kernel(const float* __restrict__ x,
                                                   const float* __restrict__ W,
                                                   const float* __restrict__ b,
                                                   float* __restrict__ h) {
    __shared__ float sW[FIN * DIM];
    __shared__ float sb[DIM];
    int tid = threadIdx.x;
    for (int t = tid; t < FIN * DIM; t += 256) sW[t] = W[t];
    if (tid < DIM) sb[tid] = b[tid];
    __syncthreads();
    int idx = blockIdx.x * 256 + tid;
    if (idx < N_NODES * DIM) {
        int i = idx >> 6, d = idx & 63;
        const float* xi = x + i * FIN;
        float acc = sb[d];
#pragma unroll
        for (int k = 0; k < FIN; ++k) acc += xi[k] * sW[k * DIM + d];
        h[idx] = acc;
    }
}

// agg[dst] += relu(h[src] + edge_attr @ We + be)
__global__ __launch_bounds__(256) void edge_kernel(const float* __restrict__ h,
                                                   const float* __restrict__ edge_attr,
                                                   const int* __restrict__ srcI,
                                                   const int* __restrict__ dstI,
                                                   const float* __restrict__ We,
                                                   const float* __restrict__ be,
                                                   float* __restrict__ agg) {
    __shared__ float sWe[FE * DIM];
    __shared__ float sbe[DIM];
    int tid = threadIdx.x;
    for (int t = tid; t < FE * DIM; t += 256) sWe[t] = We[t];
    if (tid < DIM) sbe[tid] = be[tid];
    __syncthreads();
    int e = blockIdx.x * 4 + (tid >> 6);
    int d = tid & 63;
    if (e < N_EDGES) {
        const float* ea = edge_attr + e * FE;
        float acc = sbe[d];
#pragma unroll
        for (int k = 0; k < FE; ++k) acc += ea[k] * sWe[k * DIM + d];
        int s = srcI[e], dn = dstI[e];
        float m = fmaxf(h[s * DIM + d] + acc, 0.0f);
        unsafeAtomicAdd(&agg[dn * DIM + d], m);
    }
}

// out = pooled @ W_jk + b_jk    ([S,64] @ [64,64])
__global__ __launch_bounds__(256) void decode_kernel(const float* __restrict__ pooled,
                                                     const float* __restrict__ W,
                                                     const float* __restrict__ b,
                                                     float* __restrict__ out) {
    __shared__ float sW[DIM * DIM];
    int tid = threadIdx.x;
    for (int t = tid; t < DIM * DIM; t += 256) sW[t] = W[t];
    __syncthreads();
    int idx = blockIdx.x * 256 + tid;
    if (idx < N_SUB * DIM) {
        int s = idx >> 6, d = idx & 63;
        const float* ps = pooled + s * DIM;
        float acc = b[d];
#pragma unroll
        for (int k = 0; k < DIM; ++k) acc += ps[k] * sW[k * DIM + d];
        out[idx] = acc;
    }
}

// ---------------------------------------------------------------------------
// fused per-layer kernel: 4 branches of MLP+LN on 16-node wave tiles via
// V_WMMA_F32_16X16X4_F32.  Weights are staged in LDS k-pair-interleaved so
// every A/B fragment is a single aligned 8-byte LDS load (no v_mov shuffles).
// ---------------------------------------------------------------------------
struct Branch {
    const float *W1, *b1, *W2, *b2, *g, *bn, *eps;
};
struct LayerArgs { Branch br[4]; };

// one 16x64 @ 64x64 GEMM on a wave32.
// A: LDS tile [16][TS] row-major.  Wp: LDS, k-pair interleaved v2f:
//   Wp[(k>>1)*64 + n] = { W[k][n], W[k+1][n] }  (k even)
__device__ __forceinline__ void gemm_tile(const float* __restrict__ A,
                                          const v2f* __restrict__ Wp,
                                          const float* __restrict__ bias,
                                          float* __restrict__ out,
                                          int lane, bool relu) {
    const int m  = lane & 15;          // A row / B,D column-in-tile
    const int kh = (lane >> 4) << 1;   // 0 for lanes 0-15, 2 for lanes 16-31
    v8f c[4];
    const v8f vzero = {0.f, 0.f, 0.f, 0.f, 0.f, 0.f, 0.f, 0.f};
#pragma unroll
    for (int nt = 0; nt < 4; ++nt) c[nt] = vzero;
#pragma unroll
    for (int k0 = 0; k0 < DIM; k0 += 4) {
        v2f a = *(const v2f*)&A[m * TS + k0 + kh];           // single b64 load
        const v2f* wrow = Wp + (((k0 + kh) >> 1) * DIM + m); // lane's column
#pragma unroll
        for (int nt = 0; nt < 4; ++nt) {
            c[nt] = __builtin_amdgcn_wmma_f32_16x16x4_f32(
                        false, a, false, wrow[nt * 16], (short)0, c[nt],
                        false, false);
        }
    }
    const int mh = (lane >> 4) * 8;    // C/D: vgpr r holds rows r and r+8
#pragma unroll
    for (int nt = 0; nt < 4; ++nt) {
#pragma unroll
        for (int r = 0; r < 8; ++r) {
            int col = nt * 16 + m;
            float v = c[nt][r] + bias[col];
            out[(mh + r) * TS + col] = relu ? fmaxf(v, 0.0f) : v;
        }
    }
}

__global__ __launch_bounds__(128)
void layer_kernel(const float* __restrict__ h,
                  float* __restrict__ h_next,
                  const float* __restrict__ agg,
                  const float* __restrict__ gsum,
                  const float* __restrict__ cnt,
                  const int* __restrict__ node_idx,
                  const int* __restrict__ root_idx,
                  const int* __restrict__ transpose_idx,
                  LayerArgs la) {
    __shared__ v2f  sW1[DIM * DIM / 2];     // k-pair interleaved
    __shared__ v2f  sW2[DIM * DIM / 2];
    __shared__ float sb1[DIM], sb2[DIM], sg[DIM], sbn[DIM];
    __shared__ float tile[4][2][16 * TS];   // per-wave ping/pong tiles
    __shared__ int   grow[4][16];           // per-wave gather row of "other"
    __shared__ float gscl[4][16];           // per-wave scale of "other"

    const int tid   = threadIdx.x;
    const int wv    = tid >> 5;
    const int lane  = tid & 31;
    const int node0 = blockIdx.x * 64 + wv * 16;
    const bool active = node0 < N_NODES;    // N % 16 == 0 -> tiles always full

    const int row = lane & 15;
    const int c0  = (lane >> 4) * 32;

    float accv[32];
#pragma unroll
    for (int j = 0; j < 32; ++j) accv[j] = 0.0f;

    for (int br = 0; br < 4; ++br) {
        __syncthreads();                    // previous branch done with weights
        const Branch& B = la.br[br];
        float* w1f = (float*)sW1;
        float* w2f = (float*)sW2;
        for (int t = tid; t < DIM * DIM; t += 128) {
            int k = t >> 6, n = t & 63;
            int di = ((k >> 1) * DIM + n) * 2 + (k & 1);   // pair-interleave
            w1f[di] = B.W1[t];
            w2f[di] = B.W2[t];
        }
        if (tid < DIM) { sb1[tid] = B.b1[tid]; sb2[tid] = B.b2[tid];
                         sg[tid]  = B.g[tid];  sbn[tid] = B.bn[tid]; }
        __syncthreads();

        if (active) {                       // wave-uniform -> EXEC all ones
            const float epsv = 1.0f + *B.eps;
            float* tA = tile[wv][0];
            float* tB = tile[wv][1];

            // per-row gather metadata (once per row, not per element)
            if (lane < 16) {
                int node = node0 + lane;
                int gr; float sc = 1.0f;
                if (br == 0)      gr = node;                         // agg
                else if (br == 1) gr = root_idx[node_idx[node]];     // h
                else if (br == 2) gr = transpose_idx[node];          // h
                else { gr = node_idx[node];                          // gsum/cnt
                       sc = 1.0f / fmaxf(cnt[gr], 1.0f); }
                grow[wv][lane] = gr;
                gscl[wv][lane] = sc;
            }
            const float* osrc = (br == 0) ? agg : ((br == 3) ? gsum : h);

            // build branch input tile: (1+eps)*h + scale*other   (float4 I/O)
            for (int t = lane; t < 16 * (DIM / 4); t += 32) {
                int r = t >> 4, c4 = t & 15;
                int node = node0 + r;
                v4f hv = *(const v4f*)&h[node * DIM + c4 * 4];
                v4f ov = *(const v4f*)&osrc[grow[wv][r] * DIM + c4 * 4];
                float sc = gscl[wv][r];
                v4f pre;
                pre.x = epsv * hv.x + sc * ov.x;
                pre.y = epsv * hv.y + sc * ov.y;
                pre.z = epsv * hv.z + sc * ov.z;
                pre.w = epsv * hv.w + sc * ov.w;
                *(v4f*)&tA[r * TS + c4 * 4] = pre;
            }

            gemm_tile(tA, sW1, sb1, tB, lane, true);   // y1 = relu(pre@W1+b1)
            gemm_tile(tB, sW2, sb2, tA, lane, false);  // y2 = y1@W2+b2

            // layernorm over the 64 features of each row, accumulate branch
            float s = 0.0f, ss = 0.0f;
#pragma unroll
            for (int jj = 0; jj < 8; ++jj) {
                v4f v = *(const v4f*)&tA[row * TS + c0 + jj * 4];
                s  += v.x + v.y + v.z + v.w;
                ss += v.x * v.x + v.y * v.y + v.z * v.z + v.w * v.w;
            }
            s  += __shfl_xor(s, 16, 32);
            ss += __shfl_xor(ss, 16, 32);
            float mu   = s * (1.0f / 64.0f);
            float var  = ss * (1.0f / 64.0f) - mu * mu;
            float rstd = rsqrtf(var + 1e-5f);
#pragma unroll
            for (int jj = 0; jj < 8; ++jj) {
                v4f v = *(const v4f*)&tA[row * TS + c0 + jj * 4];
                v4f g = *(const v4f*)&sg[c0 + jj * 4];
                v4f b = *(const v4f*)&sbn[c0 + jj * 4];
                accv[jj * 4 + 0] += (v.x - mu) * rstd * g.x + b.x;
                accv[jj * 4 + 1] += (v.y - mu) * rstd * g.y + b.y;
                accv[jj * 4 + 2] += (v.z - mu) * rstd * g.z + b.z;
                accv[jj * 4 + 3] += (v.w - mu) * rstd * g.w + b.w;
            }
        }
    }
    // coalesced float4 writeback of the 4-branch sum via LDS
    if (active) {
        float* tA = tile[wv][0];
#pragma unroll
        for (int jj = 0; jj < 8; ++jj) {
            v4f v; v.x = accv[jj * 4 + 0]; v.y = accv[jj * 4 + 1];
                   v.z = accv[jj * 4 + 2]; v.w = accv[jj * 4 + 3];
            *(v4f*)&tA[row * TS + c0 + jj * 4] = v;
        }
        for (int t = lane; t < 16 * (DIM / 4); t += 32) {
            int r = t >> 4, c4 = t & 15;
            *(v4f*)&h_next[(node0 + r) * DIM + c4 * 4] =
                *(const v4f*)&tA[r * TS + c4 * 4];
        }
    }
}

// ---------------------------------------------------------------------------
// host launcher
// ---------------------------------------------------------------------------
extern "C" void kernel_launch(void* const* d_in, const int* in_sizes, int n_in,
                              void* d_out, int out_size, void* d_ws, size_t ws_size,
                              hipStream_t stream) {
    (void)in_sizes; (void)n_in; (void)out_size; (void)ws_size;
    const float* x             = (const float*)d_in[0];
    const float* edge_attr     = (const float*)d_in[1];
    const int*   edge_index    = (const int*)d_in[2];
    const int*   node_idx      = (const int*)d_in[3];
    const int*   root_idx      = (const int*)d_in[4];
    const int*   transpose_idx = (const int*)d_in[5];
    const float* W_init        = (const float*)d_in[6];
    const float* b_init        = (const float*)d_in[7];
    const float* W_jk          = (const float*)d_in[38];
    const float* b_jk          = (const float*)d_in[39];
    float* out = (float*)d_out;

    char* ws = (char*)d_ws;
    const unsigned long long HB = (unsigned long long)N_NODES * DIM * sizeof(float); // 64 MB
    float* h      = (float*)(ws);
    float* h_next = (float*)(ws + HB);
    float* agg    = (float*)(ws + 2 * HB);
    float* gsum   = (float*)(ws + 3 * HB);
    float* cnt    = (float*)(ws + 3 * HB + (unsigned long long)N_SUB * DIM * sizeof(float));
    float* pooled = (float*)(ws + 3 * HB + (unsigned long long)N_SUB * DIM * sizeof(float) + 4096);

    const int* srcI = edge_index;            // edge_index[0,:]
    const int* dstI = edge_index + N_EDGES;  // edge_index[1,:]

    // subgraph member counts
    zero_kernel<<<64, 256, 0, stream>>>(cnt, (long long)N_SUB);
    count_kernel<<<(N_NODES + 255) / 256, 256, 0, stream>>>(node_idx, cnt);
    // init encoder
    init_kernel<<<(N_NODES * DIM + 255) / 256, 256, 0, stream>>>(x, W_init, b_init, h);

    for (int l = 0; l < NLAYER; ++l) {
        // lu aggregation
        zero_kernel<<<2048, 256, 0, stream>>>(agg, (long long)N_NODES * DIM);
        edge_kernel<<<(N_EDGES + 3) / 4, 256, 0, stream>>>(
            h, edge_attr, srcI, dstI,
            (const float*)d_in[8] + l * FE * DIM,   // lu_We[l]
            (const float*)d_in[9] + l * DIM,        // lu_be[l]
            agg);
        // global subgraph sums
        zero_kernel<<<256, 256, 0, stream>>>(gsum, (long long)N_SUB * DIM);
        segsum_kernel<<<(N_NODES * DIM + 255) / 256, 256, 0, stream>>>(h, node_idx, gsum);

        LayerArgs la;
        auto setB = [&](int bi, int eps_ix) {   // eps, W1, b1, W2, b2, g, bn
            la.br[bi].eps = (const float*)d_in[eps_ix]     + l;
            la.br[bi].W1  = (const float*)d_in[eps_ix + 1] + l * DIM * DIM;
            la.br[bi].b1  = (const float*)d_in[eps_ix + 2] + l * DIM;
            la.br[bi].W2  = (const float*)d_in[eps_ix + 3] + l * DIM * DIM;
            la.br[bi].b2  = (const float*)d_in[eps_ix + 4] + l * DIM;
            la.br[bi].g   = (const float*)d_in[eps_ix + 5] + l * DIM;
            la.br[bi].bn  = (const float*)d_in[eps_ix + 6] + l * DIM;
        };
        setB(0, 10);  // lu
        setB(1, 17);  // vv
        setB(2, 24);  // vu
        setB(3, 31);  // gl

        layer_kernel<<<(N_NODES + 63) / 64, 128, 0, stream>>>(
            h, h_next, agg, gsum, cnt, node_idx, root_idx, transpose_idx, la);

        float* tmp = h; h = h_next; h_next = tmp;
    }

    // SV pooling + JK decoder
    zero_kernel<<<256, 256, 0, stream>>>(pooled, (long long)N_SUB * DIM);
    segsum_kernel<<<(N_NODES * DIM + 255) / 256, 256, 0, stream>>>(h, node_idx, pooled);
    decode_kernel<<<(N_SUB * DIM + 255) / 256, 256, 0, stream>>>(pooled, W_jk, b_jk, out);
}